// SH_Msg_27384711479756
// MI455X (gfx1250) — compile-verified
//
#include <hip/hip_runtime.h>

typedef __attribute__((ext_vector_type(2))) float v2f;
typedef __attribute__((ext_vector_type(4))) float v4f;
typedef __attribute__((ext_vector_type(8))) float v8f;

#define EDGES_PER_BLOCK 256
#define TILE_STRIDE 28   // 25 features padded to 28 floats (112B, 16B-aligned rows)

// degree l occupies SH indices [l*l, (l+1)*(l+1)) -> seg(k) = floor(sqrt(k))
__device__ __forceinline__ int seg_of(int k) {
    return (k >= 16) ? 4 : (k >= 9) ? 3 : (k >= 4) ? 2 : (k >= 1) ? 1 : 0;
}

__global__ __launch_bounds__(256) void sh_edge_wmma_kernel(
    const long long* __restrict__ edge_index,  // [2*E] (row then col)
    const float*     __restrict__ node_sh,     // [N*25]
    float*           __restrict__ out,         // [E*5]
    long long E)
{
    __shared__ float tile[EDGES_PER_BLOCK * TILE_STRIDE];

    const int t = threadIdx.x;
    const long long blockBase = (long long)blockIdx.x * EDGES_PER_BLOCK;

    // ---------------- Phase 1: gather + multiply, stage products in LDS ----------------
    {
        long long e  = blockBase + t;
        long long eg = (e < E) ? e : (E - 1);
        // NT loads: keep the streaming index array from evicting the L2-resident node table
        long long r = __builtin_nontemporal_load(&edge_index[eg]);
        long long c = __builtin_nontemporal_load(&edge_index[E + eg]);
        const float* __restrict__ pa = node_sh + r * 25;
        const float* __restrict__ pb = node_sh + c * 25;
        float p[28];
#pragma unroll
        for (int k = 0; k < 25; ++k) p[k] = pa[k] * pb[k];
        p[25] = 0.0f; p[26] = 0.0f; p[27] = 0.0f;

        v4f* dst = (v4f*)&tile[t * TILE_STRIDE];   // 16B-aligned (112B row stride)
#pragma unroll
        for (int q = 0; q < 7; ++q) {
            v4f v = { p[4*q+0], p[4*q+1], p[4*q+2], p[4*q+3] };
            dst[q] = v;                            // ds_store_b128
        }
    }
    __syncthreads();

    // ---------------- Projector B-matrix chunks as register constants ----------------
    // B is 4x16 (KxN) per chunk; mirrors the documented f32 A layout:
    //   VGPR0: lanes0-15 K=0 / lanes16-31 K=2 ; VGPR1: K=1 / K=3
    const int lane  = t & 31;
    const int half  = lane >> 4;    // 0 -> lanes 0-15, 1 -> lanes 16-31
    const int n     = lane & 15;    // B column = output degree; also A row = edge-in-group
    const int khalf = half * 2;     // K offset within 4-chunk handled by this half-wave
    v2f bmat[7];
#pragma unroll
    for (int cch = 0; cch < 7; ++cch) {
        int kb = 4 * cch + khalf;
        float bx = (kb     < 25 && seg_of(kb)     == n) ? 1.0f : 0.0f;
        float by = (kb + 1 < 25 && seg_of(kb + 1) == n) ? 1.0f : 0.0f;
        v2f b; b.x = bx; b.y = by;
        bmat[cch] = b;
    }

    // ---------------- Phase 2: WMMA projection (16 edges per group, 2 groups/wave) ----
    const int wave = t >> 5;
#pragma unroll
    for (int g = 0; g < 2; ++g) {
        const int el = wave * 32 + g * 16 + n;   // A-matrix row M = lane&15 -> local edge
        v8f d = {0.f, 0.f, 0.f, 0.f, 0.f, 0.f, 0.f, 0.f};
#pragma unroll
        for (int cch = 0; cch < 7; ++cch) {
            // (el*28 + 4c + khalf) is even -> 8B-aligned ds_load_b64 (fuses to 2addr)
            v2f a = *(const v2f*)&tile[el * TILE_STRIDE + 4 * cch + khalf];
            d = __builtin_amdgcn_wmma_f32_16x16x4_f32(
                    false, a, false, bmat[cch], (short)0, d, false, false);
        }
        // D layout: VGPR r, lanes0-15 -> (M=r, N=lane); lanes16-31 -> (M=8+r, N=lane-16)
        // This lane holds degree `n` of edges e0..e0+7; addresses are base + r*20B.
        const int deg  = n;
        const int mofs = half * 8;
        if (deg < 5) {
            const long long e0 = blockBase + wave * 32 + g * 16 + mofs;
            float* po = out + e0 * 5 + deg;
            if (e0 + 7 < E) {
                // Fast path (all full blocks): 8 NT stores with immediate offsets
#pragma unroll
                for (int r = 0; r < 8; ++r)
                    __builtin_nontemporal_store(d[r], po + r * 5);
            } else {
#pragma unroll
                for (int r = 0; r < 8; ++r)
                    if (e0 + r < E)
                        __builtin_nontemporal_store(d[r], po + r * 5);
            }
        }
    }
}

extern "C" void kernel_launch(void* const* d_in, const int* in_sizes, int n_in,
                              void* d_out, int out_size, void* d_ws, size_t ws_size,
                              hipStream_t stream) {
    const long long* edge_index = (const long long*)d_in[0]; // int64 per reference
    const float*     node_sh    = (const float*)d_in[1];
    float*           out        = (float*)d_out;

    const long long E = (long long)in_sizes[0] / 2;
    const int blocks  = (int)((E + EDGES_PER_BLOCK - 1) / EDGES_PER_BLOCK);

    sh_edge_wmma_kernel<<<blocks, 256, 0, stream>>>(edge_index, node_sh, out, E);
}